// ProbUnifMPS_38113539784979
// MI455X (gfx1250) — compile-verified
//
#include <hip/hip_runtime.h>
#include <math.h>

// ProbUnifMPS for MI455X (gfx1250, wave32).
// fp32 WMMA (v_wmma_f32_16x16x4_f32) for all 128x128x128 matmuls.
// B=32, S=128, V=64, D=128.
//
// Convention: the B operand of every matmul is stored TRANSPOSED in LDS
// (Bt[n*PITCH+k] = B[k][n]) so both A and B fragments are contiguous 2-dword
// LDS reads (single ds_load_2addr_b32 into an adjacent VGPR pair, no repack).

typedef float v2f __attribute__((ext_vector_type(2)));
typedef float v8f __attribute__((ext_vector_type(8)));

#define D_DIM   128
#define PITCH   129              // odd pitch: lane-stride-PITCH LDS access is bank-conflict free
#define MATF    16384            // floats per 128x128 matrix
#define NB      32               // batch
#define NV      64               // vocab
#define LDSMAT  (D_DIM * PITCH)  // floats per LDS matrix

// ---------------------------------------------------------------------------
// WMMA 128x128x128 matmul: 256 threads = 8 waves; wave w computes the 16-row
// strip [16w,16w+16) x 128 of D = A*B, accumulating in 8 v8f tiles.
// A row-major (pitch PITCH); Bt transposed (Bt[n*PITCH+k] = B[k][n]).
// Fragment layouts per CDNA5 ISA (16x4 A / 4x16 B / 16x16 C-D, f32):
//   A: lanes 0-15 row M=m0+l: v0=K0,v1=K1 ; lanes 16-31 same rows: v0=K2,v1=K3
//   B: lanes 0-15 col N=n0+l: v0=row K0, v1=row K1 ; lanes 16-31: rows K2,K3
//   C/D: lane l<16 holds D[m0+r][n0+l] in vgpr r; lanes 16-31 rows m0+8+r.
// ---------------------------------------------------------------------------
__device__ __forceinline__ void wmma_matmul_strip(const float* __restrict__ A,
                                                  const float* __restrict__ Bt,
                                                  v8f acc[8]) {
  const int tid  = threadIdx.x;
  const int w    = tid >> 5;
  const int lane = tid & 31;
  const int l    = lane & 15;
  const int h    = (lane >> 4) << 1;   // 0 for lanes 0-15, 2 for lanes 16-31
  const float* arow = A + (w * 16 + l) * PITCH + h;   // this wave's A row base
  const float* bcol = Bt + l * PITCH + h;             // n-tile 0 column base

  for (int k0 = 0; k0 < D_DIM; k0 += 4) {
    v2f a;
    a[0] = arow[k0];
    a[1] = arow[k0 + 1];
#pragma unroll
    for (int n = 0; n < 8; ++n) {
      const float* bp = bcol + n * (16 * PITCH) + k0;
      v2f b;
      b[0] = bp[0];
      b[1] = bp[1];
      acc[n] = __builtin_amdgcn_wmma_f32_16x16x4_f32(
          /*neg_a=*/false, a, /*neg_b=*/false, b,
          /*c_mod=*/(short)0, acc[n], /*reuse_a=*/false, /*reuse_b=*/false);
    }
  }
}

__device__ __forceinline__ void zero_acc(v8f acc[8]) {
#pragma unroll
  for (int n = 0; n < 8; ++n) acc[n] = (v8f){0.f, 0.f, 0.f, 0.f, 0.f, 0.f, 0.f, 0.f};
}

// Store this wave's strip row-major with scale; dst row pitch `pitch` (floats).
__device__ __forceinline__ void store_strip(float* dst, int pitch,
                                            const v8f acc[8], float scale) {
  const int tid  = threadIdx.x;
  const int w    = tid >> 5;
  const int lane = tid & 31;
  const int l    = lane & 15;
  const int m0   = w * 16 + ((lane >> 4) << 3);  // +8 rows for hi half-wave
#pragma unroll
  for (int n = 0; n < 8; ++n)
#pragma unroll
    for (int r = 0; r < 8; ++r)
      dst[(size_t)(m0 + r) * pitch + n * 16 + l] = acc[n][r] * scale;
}

// Store this wave's strip TRANSPOSED into LDS: dst[n*PITCH+m] = D[m][n]*scale.
// Per-lane writes are contiguous in m (r), lane-stride PITCH in n.
__device__ __forceinline__ void store_strip_t(float* dst, const v8f acc[8],
                                              float scale) {
  const int tid  = threadIdx.x;
  const int w    = tid >> 5;
  const int lane = tid & 31;
  const int l    = lane & 15;
  const int m0   = w * 16 + ((lane >> 4) << 3);
#pragma unroll
  for (int n = 0; n < 8; ++n) {
    float* col = dst + (size_t)(n * 16 + l) * PITCH + m0;
#pragma unroll
    for (int r = 0; r < 8; ++r) col[r] = acc[n][r] * scale;
  }
}

// Deterministic block-wide Frobenius norm of the full 128x128 result
// distributed across all 256 threads' accumulators. red: shared float[256+].
__device__ __forceinline__ float block_frob(const v8f acc[8], float* red) {
  float s = 0.f;
#pragma unroll
  for (int n = 0; n < 8; ++n)
#pragma unroll
    for (int r = 0; r < 8; ++r) s += acc[n][r] * acc[n][r];
  const int tid = threadIdx.x;
  __syncthreads();               // protect red reuse + fence all LDS reads of matmul
  red[tid] = s;
  __syncthreads();
  for (int st = 128; st > 0; st >>= 1) {
    if (tid < st) red[tid] += red[tid + st];
    __syncthreads();
  }
  return sqrtf(red[0]);
}

// Cooperative load of a row-major 128x128 matrix into LDS (pitch PITCH), scaled.
__device__ __forceinline__ void load_mat_lds(float* dst, const float* __restrict__ src,
                                             float scale) {
  for (int e = threadIdx.x; e < MATF; e += 256)
    dst[(e >> 7) * PITCH + (e & 127)] = src[e] * scale;
}

// Cooperative TRANSPOSED load: dst[c*PITCH+r] = src[r*128+c] * scale.
// Global reads coalesced; LDS writes lane-stride PITCH (conflict-free).
__device__ __forceinline__ void load_mat_lds_t(float* dst, const float* __restrict__ src,
                                               float scale) {
  for (int e = threadIdx.x; e < MATF; e += 256)
    dst[(e & 127) * PITCH + (e >> 7)] = src[e] * scale;
}

// One-shot inter-workgroup barrier (counter pre-zeroed per launch).
__device__ __forceinline__ void gbar(int* cnt, int idx, int target) {
  __threadfence();
  __syncthreads();
  if (threadIdx.x == 0) {
    __hip_atomic_fetch_add(&cnt[idx], 1, __ATOMIC_ACQ_REL, __HIP_MEMORY_SCOPE_AGENT);
    while (__hip_atomic_load(&cnt[idx], __ATOMIC_ACQUIRE, __HIP_MEMORY_SCOPE_AGENT) < target)
      __builtin_amdgcn_s_sleep(2);
  }
  __syncthreads();
  __threadfence();
}

// ---------------------------------------------------------------------------
// Kernel: init — zero barrier counters, build M0 = outer(alpha, alpha).
// grid 64 x 256
// ---------------------------------------------------------------------------
__global__ void k_init(const float* __restrict__ ev, float* __restrict__ mraw,
                       int* __restrict__ counters) {
  const int e = blockIdx.x * 256 + threadIdx.x;
  if (e < MATF) mraw[e] = ev[e >> 7] * ev[e & 127];  // alpha = ev[0:128]
  if (e < 256) counters[e] = 0;
}

// ---------------------------------------------------------------------------
// Kernel: fused tree levels 0+1. grid = B*32 WGs of 256 threads.
// WG (b,l) consumes tokens s=4l..4l+3:  P1=C[i1]@C[i0], P2=C[i3]@C[i2],
// Q = (P2/n2)@(P1/n1); writes Q/n3 and log n1+log n2+log n3.
// ---------------------------------------------------------------------------
__global__ void k_level01(const int* __restrict__ idx, const float* __restrict__ core,
                          float* __restrict__ out, float* __restrict__ lognode) {
  extern __shared__ float sm[];
  __shared__ float red[257];
  float* bA = sm;                 // A operand, row-major
  float* bB = sm + LDSMAT;        // B operand, transposed
  float* bP = sm + 2 * LDSMAT;    // P1' kept transposed (B of final matmul)

  const int b = blockIdx.x >> 5;
  const int l = blockIdx.x & 31;
  const int* ip = idx + b * 128 + 4 * l;
  const int i0 = ip[0], i1 = ip[1], i2 = ip[2], i3 = ip[3];

  load_mat_lds(bA, core + (size_t)i1 * MATF, 1.f);
  load_mat_lds_t(bB, core + (size_t)i0 * MATF, 1.f);
  __syncthreads();

  v8f acc[8];
  zero_acc(acc);
  wmma_matmul_strip(bA, bB, acc);                 // P1 = C[i1] @ C[i0]
  const float n1 = block_frob(acc, red);
  store_strip_t(bP, acc, 1.f / n1);               // P1' (transposed) -> bP
  __syncthreads();

  load_mat_lds(bA, core + (size_t)i3 * MATF, 1.f);
  load_mat_lds_t(bB, core + (size_t)i2 * MATF, 1.f);
  __syncthreads();

  zero_acc(acc);
  wmma_matmul_strip(bA, bB, acc);                 // P2 = C[i3] @ C[i2]
  const float n2 = block_frob(acc, red);          // (fences all LDS reads)
  store_strip(bA, PITCH, acc, 1.f / n2);          // P2' row-major -> bA
  __syncthreads();

  zero_acc(acc);
  wmma_matmul_strip(bA, bP, acc);                 // Q = P2' @ P1'
  const float n3 = block_frob(acc, red);
  store_strip(out + ((size_t)b * 32 + l) * MATF, D_DIM, acc, 1.f / n3);
  if (threadIdx.x == 0)
    lognode[b * 63 + l] = logf(n1) + logf(n2) + logf(n3);
}

// ---------------------------------------------------------------------------
// Kernel: generic tree pair level. grid = B*count_out WGs.
// prod[l] = in[2l+1] @ in[2l], normalized; log(n) -> lognode[b*63+lbase+l].
// ---------------------------------------------------------------------------
__global__ void k_pair(const float* __restrict__ in, float* __restrict__ out,
                       float* __restrict__ lognode, int count_out, int lbase) {
  extern __shared__ float sm[];
  __shared__ float red[257];
  float* bA = sm;
  float* bB = sm + LDSMAT;

  const int b  = blockIdx.x / count_out;
  const int l  = blockIdx.x % count_out;
  const int ci = 2 * count_out;

  load_mat_lds(bA, in + ((size_t)b * ci + 2 * l + 1) * MATF, 1.f);
  load_mat_lds_t(bB, in + ((size_t)b * ci + 2 * l) * MATF, 1.f);
  __syncthreads();

  v8f acc[8];
  zero_acc(acc);
  wmma_matmul_strip(bA, bB, acc);
  const float n = block_frob(acc, red);
  store_strip(out + ((size_t)b * count_out + l) * MATF, D_DIM, acc, 1.f / n);
  if (threadIdx.x == 0) lognode[b * 63 + lbase + l] = logf(n);
}

// ---------------------------------------------------------------------------
// Kernel: log_norm — persistent cooperative kernel, 64 WGs (one per C_i),
// 128 sequential steps of M <- sum_i C_i M C_i^T with per-step Frobenius
// normalization folded into the next step's load. C_i stays in LDS for the
// whole kernel (row-major cbuf serves BOTH as Bt of matmul-1, since
// Bt[n][k] = (C^T)[k][n] = C[n][k], AND as A of matmul-2).
// Two global barriers per step via one-shot counters.
// ---------------------------------------------------------------------------
__global__ void k_lognorm(const float* __restrict__ core, const float* __restrict__ ev,
                          float* __restrict__ partials, float* __restrict__ mraw,
                          float* __restrict__ wgsumsq, int* __restrict__ counters,
                          float* __restrict__ lognorm_out) {
  extern __shared__ float sm[];
  __shared__ float red[257];
  float* cbuf = sm;               // C_i row-major, resident all steps
  float* mbuf = sm + LDSMAT;      // M (A of matmul-1), row-major
  float* tbuf = sm + 2 * LDSMAT;  // T, stored transposed (B of matmul-2)

  const int i   = blockIdx.x;    // 0..63
  const int tid = threadIdx.x;

  load_mat_lds(cbuf, core + (size_t)i * MATF, 1.f);

  float acc_log = 0.f;
  for (int s = 0; s < 128; ++s) {
    // ---- phase A: load normalized M into LDS (scale from previous step) ----
    float scale = 1.f;
    if (s > 0) {
      float ssq = 0.f;
      for (int p = 0; p < NV; ++p) ssq += wgsumsq[p];   // deterministic, redundant
      const float n = sqrtf(ssq);
      scale = 1.f / n;
      acc_log += logf(n);
    }
    __syncthreads();                 // mbuf WAR vs previous step's matmul reads
    load_mat_lds(mbuf, mraw, scale);
    __syncthreads();

    // ---- T = M @ C_i^T  (Bt := cbuf) ----
    v8f a1[8];
    zero_acc(a1);
    wmma_matmul_strip(mbuf, cbuf, a1);
    store_strip_t(tbuf, a1, 1.f);        // per-wave disjoint columns
    __syncthreads();

    // ---- G_i = C_i @ T  (A := cbuf, Bt := tbuf) ----
    v8f a2[8];
    zero_acc(a2);
    wmma_matmul_strip(cbuf, tbuf, a2);
    store_strip(partials + (size_t)i * MATF, D_DIM, a2, 1.f);

    gbar(counters, 2 * s, NV);

    // ---- reduce: WG i owns flat elements [i*256, i*256+256) ----
    {
      const int e = i * 256 + tid;
      float v = 0.f;
      for (int p = 0; p < NV; ++p) v += partials[(size_t)p * MATF + e];
      mraw[e] = v;
      const float sq = v * v;
      __syncthreads();
      red[tid] = sq;
      __syncthreads();
      for (int st = 128; st > 0; st >>= 1) {
        if (tid < st) red[tid] += red[tid + st];
        __syncthreads();
      }
      if (tid == 0) wgsumsq[i] = red[0];
    }

    gbar(counters, 2 * s + 1, NV);
  }

  // ---- finish: last norm + z = omega^T (M/n_last) omega ----
  float ssq = 0.f;
  for (int p = 0; p < NV; ++p) ssq += wgsumsq[p];
  const float nlast = sqrtf(ssq);
  acc_log += logf(nlast);

  if (i == 0) {
    const float* omega = ev + 128;
    float sv = 0.f;
    if (tid < 128) {
      float rowd = 0.f;
      for (int j = 0; j < 128; ++j) rowd += mraw[tid * 128 + j] * omega[j];
      sv = omega[tid] * rowd;
    }
    __syncthreads();
    red[tid] = sv;
    __syncthreads();
    for (int st = 128; st > 0; st >>= 1) {
      if (tid < st) red[tid] += red[tid + st];
      __syncthreads();
    }
    if (tid == 0) lognorm_out[0] = logf(red[0] / nlast) + acc_log;
  }
}

// ---------------------------------------------------------------------------
// Kernel: final — psi[b] = omega^T M_root[b] alpha; out = 2*(log|psi| +
// sum(lognode[b])) - log_norm.  grid = 32 WGs x 128 threads.
// ---------------------------------------------------------------------------
__global__ void k_final(const float* __restrict__ roots, const float* __restrict__ ev,
                        const float* __restrict__ lognode,
                        const float* __restrict__ lognorm_s, float* __restrict__ out) {
  __shared__ float red[128];
  const int b = blockIdx.x;
  const int tid = threadIdx.x;
  const float* M = roots + (size_t)b * MATF;
  const float* alpha = ev;
  const float* omega = ev + 128;

  float rowd = 0.f;
  for (int j = 0; j < 128; ++j) rowd += M[tid * 128 + j] * alpha[j];
  red[tid] = omega[tid] * rowd;
  __syncthreads();
  for (int st = 64; st > 0; st >>= 1) {
    if (tid < st) red[tid] += red[tid + st];
    __syncthreads();
  }
  if (tid == 0) {
    float ls = 0.f;
    for (int k = 0; k < 63; ++k) ls += lognode[b * 63 + k];
    out[b] = 2.f * (logf(fabsf(red[0])) + ls) - lognorm_s[0];
  }
}

// ---------------------------------------------------------------------------
extern "C" void kernel_launch(void* const* d_in, const int* in_sizes, int n_in,
                              void* d_out, int out_size, void* d_ws, size_t ws_size,
                              hipStream_t stream) {
  (void)in_sizes; (void)n_in; (void)out_size; (void)ws_size;
  const int*   idx  = (const int*)d_in[0];    // (32,128) int32
  const float* core = (const float*)d_in[1];  // (64,128,128) f32
  const float* ev   = (const float*)d_in[2];  // (2,128) f32
  float* out = (float*)d_out;                 // (32,) f32
  float* ws  = (float*)d_ws;

  // Workspace layout (floats). Total ~= 100.7 MB.
  float* bufA      = ws;                                  // 32*32 matrices
  float* bufB      = bufA + (size_t)NB * 32 * MATF;       // 32*16 matrices
  float* partials  = bufB + (size_t)NB * 16 * MATF;       // 64 matrices
  float* mraw      = partials + (size_t)NV * MATF;        // 1 matrix
  float* wgsq      = mraw + MATF;                         // 64
  float* lognode   = wgsq + 64;                           // 32*63
  float* lognorm_s = lognode + (size_t)NB * 63;           // 1 (+pad)
  int*   counters  = (int*)(lognorm_s + 16);              // 256 ints

  const size_t lds2 = (size_t)2 * LDSMAT * sizeof(float); // ~132 KB
  const size_t lds3 = (size_t)3 * LDSMAT * sizeof(float); // ~198 KB (<320 KB/WGP)

  k_init<<<64, 256, 0, stream>>>(ev, mraw, counters);

  // Tree reduction over the sequence (levels 0+1 fused, then 2..6 ping-pong).
  k_level01<<<NB * 32, 256, lds3, stream>>>(idx, core, bufA, lognode);
  k_pair<<<NB * 16, 256, lds2, stream>>>(bufA, bufB, lognode, 16, 32);
  k_pair<<<NB * 8,  256, lds2, stream>>>(bufB, bufA, lognode, 8, 48);
  k_pair<<<NB * 4,  256, lds2, stream>>>(bufA, bufB, lognode, 4, 56);
  k_pair<<<NB * 2,  256, lds2, stream>>>(bufB, bufA, lognode, 2, 60);
  k_pair<<<NB * 1,  256, lds2, stream>>>(bufA, bufB, lognode, 1, 62);

  // Sequential 128-step transfer-operator iteration (persistent cooperative).
  k_lognorm<<<NV, 256, lds3, stream>>>(core, ev, partials, mraw, wgsq,
                                       counters, lognorm_s);

  k_final<<<NB, 128, 0, stream>>>(bufB, ev, lognode, lognorm_s, out);
}